// NormalShader_65146063945794
// MI455X (gfx1250) — compile-verified
//
#include <hip/hip_runtime.h>

// ---------------------------------------------------------------------------
// NormalShader (SoftRas softmax-rgb-blend of interpolated vertex normals)
//
// Roofline on MI455X: ~113 MB HBM streaming traffic, ~1 GFLOP -> memory bound
// (<10 FLOP/B vs 23.3 TB/s). No dense matmul exists (per-sample 1x3 * 3x3 with
// gathered operands), so WMMA is inapplicable; the win is pure data movement:
//  - b128 coalesced loads of the K=4-contiguous per-pixel streams
//  - gfx1250 async global->LDS staging (ASYNCcnt path, confirmed present)
//  - NT temporal hints on read-once streams; RT caching for the L2-resident
//    gather tables (faces 2.4MB + normals 1.2MB << 192MB L2)
//  - LDS transpose so output is stored as coalesced NT b128
// ---------------------------------------------------------------------------

typedef float __attribute__((ext_vector_type(4))) f32x4;
typedef int   __attribute__((ext_vector_type(4))) i32x4;

#if defined(__HIP_DEVICE_COMPILE__) &&                                        \
    __has_builtin(__builtin_amdgcn_global_load_async_to_lds_b128) &&          \
    __has_builtin(__builtin_amdgcn_s_wait_asynccnt)
#define USE_ASYNC_LDS 1
// Builtin signature (from hipcc diagnostic): param0 = int4 in AS1 (global),
// param1 = int4 in AS3 (LDS), then imm offset, imm cpol.
typedef int v4i_t __attribute__((vector_size(16)));
typedef __attribute__((address_space(1))) v4i_t* g_v4i_p;
typedef __attribute__((address_space(3))) v4i_t* l_v4i_p;
#else
#define USE_ASYNC_LDS 0
#endif

#define BLOCK 256

__global__ __launch_bounds__(BLOCK)
void normal_shader_kernel(const float* __restrict__ vnrm,   // [V,3]
                          const f32x4* __restrict__ bary,   // [P,3] of f32x4 (12 f/pixel)
                          const f32x4* __restrict__ dists,  // [P]
                          const f32x4* __restrict__ zbuf,   // [P]
                          const int*   __restrict__ faces,  // [F,3]
                          const i32x4* __restrict__ p2f,    // [P]
                          float*       __restrict__ out,    // [P,3]
                          int P)
{
    constexpr float SIGMA_INV = 1.0e4f;   // 1/SIGMA
    constexpr float GAMMA_INV = 1.0e4f;   // 1/GAMMA
    constexpr float EPS       = 1e-10f;
    constexpr float ZFAR      = 100.0f;
    constexpr float ZSCALE    = 1.0f / 99.0f;   // 1/(ZFAR-ZNEAR)

    const int tid = threadIdx.x;
    const int pix = blockIdx.x * BLOCK + tid;

    f32x4 d4 = {0.f, 0.f, 0.f, 0.f};
    f32x4 z4 = {0.f, 0.f, 0.f, 0.f};
    f32x4 b0 = d4, b1 = d4, b2 = d4;
    i32x4 f4 = {-1, -1, -1, -1};

#if USE_ASYNC_LDS
    // Stage the four read-once streams through LDS with the gfx1250 async
    // copy engine (per-lane b128 transfers, tracked by ASYNCcnt).
    __shared__ __align__(16) float s_d[BLOCK * 4];
    __shared__ __align__(16) float s_z[BLOCK * 4];
    __shared__ __align__(16) int   s_f[BLOCK * 4];
    __shared__ __align__(16) float s_b[BLOCK * 12];
    if (pix < P) {
        __builtin_amdgcn_global_load_async_to_lds_b128((g_v4i_p)&dists[pix], (l_v4i_p)&s_d[tid * 4], 0, 0);
        __builtin_amdgcn_global_load_async_to_lds_b128((g_v4i_p)&zbuf[pix],  (l_v4i_p)&s_z[tid * 4], 0, 0);
        __builtin_amdgcn_global_load_async_to_lds_b128((g_v4i_p)&p2f[pix],   (l_v4i_p)&s_f[tid * 4], 0, 0);
        const long bb = (long)pix * 3;
        __builtin_amdgcn_global_load_async_to_lds_b128((g_v4i_p)&bary[bb + 0], (l_v4i_p)&s_b[tid * 12 + 0], 0, 0);
        __builtin_amdgcn_global_load_async_to_lds_b128((g_v4i_p)&bary[bb + 1], (l_v4i_p)&s_b[tid * 12 + 4], 0, 0);
        __builtin_amdgcn_global_load_async_to_lds_b128((g_v4i_p)&bary[bb + 2], (l_v4i_p)&s_b[tid * 12 + 8], 0, 0);
    }
    __builtin_amdgcn_s_wait_asynccnt(0);
    __syncthreads();
    if (pix < P) {
        d4 = *(const f32x4*)&s_d[tid * 4];
        z4 = *(const f32x4*)&s_z[tid * 4];
        f4 = *(const i32x4*)&s_f[tid * 4];
        b0 = *(const f32x4*)&s_b[tid * 12 + 0];
        b1 = *(const f32x4*)&s_b[tid * 12 + 4];
        b2 = *(const f32x4*)&s_b[tid * 12 + 8];
    }
#else
    if (pix < P) {
        d4 = __builtin_nontemporal_load(&dists[pix]);
        z4 = __builtin_nontemporal_load(&zbuf[pix]);
        f4 = __builtin_nontemporal_load(&p2f[pix]);
        const long bb = (long)pix * 3;
        b0 = __builtin_nontemporal_load(&bary[bb + 0]);
        b1 = __builtin_nontemporal_load(&bary[bb + 1]);
        b2 = __builtin_nontemporal_load(&bary[bb + 2]);
    }
#endif

    float r = 0.f, g = 0.f, b = 0.f;
    if (pix < P) {
        const float m0 = (f4.x >= 0) ? 1.f : 0.f;
        const float m1 = (f4.y >= 0) ? 1.f : 0.f;
        const float m2 = (f4.z >= 0) ? 1.f : 0.f;
        const float m3 = (f4.w >= 0) ? 1.f : 0.f;

        const float zi0 = (ZFAR - z4.x) * ZSCALE * m0;
        const float zi1 = (ZFAR - z4.y) * ZSCALE * m1;
        const float zi2 = (ZFAR - z4.z) * ZSCALE * m2;
        const float zi3 = (ZFAR - z4.w) * ZSCALE * m3;
        float zmax = fmaxf(fmaxf(zi0, zi1), fmaxf(zi2, zi3));
        zmax = fmaxf(zmax, EPS);

        // w_k = mask * sigmoid(-d/sigma) * exp((z_inv - zmax)/gamma)
        const float w0 = m0 / (1.f + expf(d4.x * SIGMA_INV)) * expf((zi0 - zmax) * GAMMA_INV);
        const float w1 = m1 / (1.f + expf(d4.y * SIGMA_INV)) * expf((zi1 - zmax) * GAMMA_INV);
        const float w2 = m2 / (1.f + expf(d4.z * SIGMA_INV)) * expf((zi2 - zmax) * GAMMA_INV);
        const float w3 = m3 / (1.f + expf(d4.w * SIGMA_INV)) * expf((zi3 - zmax) * GAMMA_INV);

        const float delta = fmaxf(expf((EPS - zmax) * GAMMA_INV), EPS);
        const float denom = w0 + w1 + w2 + w3 + delta;

        float accx = delta, accy = delta, accz = delta;   // delta * BG(1,1,1)

        // Gather+interp per valid sample; L2-resident tables -> default RT hint.
        auto sample = [&](int fi, float w, float ba, float bbw, float bc) {
            if (fi >= 0) {
                const int v0 = faces[fi * 3 + 0];
                const int v1 = faces[fi * 3 + 1];
                const int v2 = faces[fi * 3 + 2];
                const float nx = ba * vnrm[v0 * 3 + 0] + bbw * vnrm[v1 * 3 + 0] + bc * vnrm[v2 * 3 + 0];
                const float ny = ba * vnrm[v0 * 3 + 1] + bbw * vnrm[v1 * 3 + 1] + bc * vnrm[v2 * 3 + 1];
                const float nz = ba * vnrm[v0 * 3 + 2] + bbw * vnrm[v1 * 3 + 2] + bc * vnrm[v2 * 3 + 2];
                accx += w * nx;  accy += w * ny;  accz += w * nz;
            }
        };
        sample(f4.x, w0, b0.x, b0.y, b0.z);
        sample(f4.y, w1, b0.w, b1.x, b1.y);
        sample(f4.z, w2, b1.z, b1.w, b2.x);
        sample(f4.w, w3, b2.y, b2.z, b2.w);

        const float inv = 1.f / denom;
        r = accx * inv;  g = accy * inv;  b = accz * inv;
        float n = sqrtf(r * r + g * g + b * b);
        n = fmaxf(n, 1e-12f);
        const float rn = 1.f / n;
        r *= rn;  g *= rn;  b *= rn;
    }

    // LDS transpose: [lane][rgb] -> contiguous floats, then coalesced NT b128
    // stores (3072B per block in 192 lane-stores).
    __shared__ __align__(16) float s_out[BLOCK * 3];
    s_out[tid * 3 + 0] = r;
    s_out[tid * 3 + 1] = g;
    s_out[tid * 3 + 2] = b;
    __syncthreads();

    const long total = (long)P * 3;
    const long base  = (long)blockIdx.x * (BLOCK * 3);
    if (tid < (BLOCK * 3) / 4) {
        const long gi = base + (long)tid * 4;
        if (gi + 4 <= total) {
            f32x4 v = *(const f32x4*)&s_out[tid * 4];
            __builtin_nontemporal_store(v, (f32x4*)&out[gi]);
        } else {
            for (int e = 0; e < 4; ++e)
                if (gi + e < total) out[gi + e] = s_out[tid * 4 + e];
        }
    }
}

extern "C" void kernel_launch(void* const* d_in, const int* in_sizes, int n_in,
                              void* d_out, int out_size, void* d_ws, size_t ws_size,
                              hipStream_t stream) {
    // setup_inputs order: verts(0), vertex_normals(1), bary_coords(2),
    //                     dists(3), zbuf(4), faces(5), pix_to_face(6)
    const float* vnrm  = (const float*)d_in[1];
    const f32x4* bary  = (const f32x4*)d_in[2];
    const f32x4* dists = (const f32x4*)d_in[3];
    const f32x4* zbuf  = (const f32x4*)d_in[4];
    const int*   faces = (const int*)d_in[5];
    const i32x4* p2f   = (const i32x4*)d_in[6];
    float* out = (float*)d_out;

    const int P = in_sizes[3] / 4;          // N*H*W (K=4 innermost)
    const int blocks = (P + BLOCK - 1) / BLOCK;
    hipLaunchKernelGGL(normal_shader_kernel, dim3(blocks), dim3(BLOCK), 0, stream,
                       vnrm, bary, dists, zbuf, faces, p2f, out, P);
}